// GCNeXt_37649683316928
// MI455X (gfx1250) — compile-verified
//
#include <hip/hip_runtime.h>
#include <hip/hip_bf16.h>

typedef __attribute__((ext_vector_type(2))) float v2f;
typedef __attribute__((ext_vector_type(8))) float v8f;

#define BSZ 16
#define CCH 256
#define NN  2048
#define WD  128

#define USE_ASYNC 1

#if USE_ASYNC
// GVS form: per-lane LDS byte offset in VDST vgpr, per-lane 32-bit global byte
// offset in VADDR vgpr, uniform 64-bit base in SADDR sgpr pair.
#define ASYNC_STAGE(ldsoff, goff, base)                                        \
    asm volatile("global_load_async_to_lds_b32 %0, %1, %2"                     \
                 :: "v"(ldsoff), "v"(goff), "s"(base) : "memory")
#define ASYNC_WAIT_ALL()  asm volatile("s_wait_asynccnt 0x0" ::: "memory")
#define ASYNC_WAIT_PREV() asm volatile("s_wait_asynccnt 0x20" ::: "memory")
#endif

static __device__ __forceinline__ v8f wmma_f32_k4(v2f a, v2f b, v8f c) {
    return __builtin_amdgcn_wmma_f32_16x16x4_f32(false, a, false, b, (short)0, c,
                                                 false, false);
}

// ---------------------------------------------------------------------------
// sq[b,n] = sum_c x[b,c,n]^2
// ---------------------------------------------------------------------------
__global__ __launch_bounds__(256) void k_sqnorm(const float* __restrict__ x,
                                                float* __restrict__ sq) {
    int i = blockIdx.x * 256 + threadIdx.x;     // i = b*NN + n
    int b = i >> 11, n = i & (NN - 1);
    const float* xb = x + (size_t)b * CCH * NN + n;
    float s = 0.f;
    for (int c = 0; c < CCH; ++c) {
        float v = xb[(size_t)c * NN];
        s = fmaf(v, v, s);
    }
    sq[i] = s;
}

// ---------------------------------------------------------------------------
// out[b] = (relu?)(A(MDIM x KDIM) * Bm[b](KDIM x NN) + bias), row stride NN.
// One 16x64 strip per wave: one shared A fragment feeds 4 WMMAs per k-step.
// ---------------------------------------------------------------------------
template <int MDIM, int KDIM, int RELU>
__global__ __launch_bounds__(256) void k_gemm(const float* __restrict__ A,
                                              const float* __restrict__ bias,
                                              const float* __restrict__ Bm,
                                              float* __restrict__ out,
                                              int strideB, int strideO) {
    const int TN4 = NN / 64;                    // 32 n-strips
    const int TM = MDIM / 16;
    int wave = threadIdx.x >> 5, lane = threadIdx.x & 31;
    int hi = lane >> 4, ln = lane & 15;
    int tile = blockIdx.x * 8 + wave;
    int b = tile / (TM * TN4);
    int r = tile % (TM * TN4);
    int m0 = (r / TN4) * 16, n0 = (r % TN4) * 64;

    const float* Ab = A + (size_t)(m0 + ln) * KDIM + 2 * hi;
    const float* Bb = Bm + (size_t)b * strideB + n0 + ln;

    v8f acc0 = {}, acc1 = {}, acc2 = {}, acc3 = {};
    for (int k = 0; k < KDIM; k += 4) {
        v2f a = *(const v2f*)(Ab + k);          // contiguous pair -> b64
        size_t r0 = (size_t)(k + 2 * hi) * NN, r1 = r0 + NN;
        v2f b0, b1, b2, b3;
        b0.x = Bb[r0];      b0.y = Bb[r1];
        b1.x = Bb[r0 + 16]; b1.y = Bb[r1 + 16];
        b2.x = Bb[r0 + 32]; b2.y = Bb[r1 + 32];
        b3.x = Bb[r0 + 48]; b3.y = Bb[r1 + 48];
        acc0 = wmma_f32_k4(a, b0, acc0);
        acc1 = wmma_f32_k4(a, b1, acc1);
        acc2 = wmma_f32_k4(a, b2, acc2);
        acc3 = wmma_f32_k4(a, b3, acc3);
    }
    float* ob = out + (size_t)b * strideO + n0 + ln;
#pragma unroll
    for (int v = 0; v < 8; ++v) {
        int m = m0 + v + 8 * hi;
        float bm = bias[m];
        float y0 = acc0[v] + bm, y1 = acc1[v] + bm;
        float y2 = acc2[v] + bm, y3 = acc3[v] + bm;
        if (RELU) {
            y0 = fmaxf(y0, 0.f); y1 = fmaxf(y1, 0.f);
            y2 = fmaxf(y2, 0.f); y3 = fmaxf(y3, 0.f);
        }
        float* op = ob + (size_t)m * NN;
        op[0] = y0; op[16] = y1; op[32] = y2; op[48] = y3;
    }
}

// ---------------------------------------------------------------------------
// Grouped temporal conv, kernel 3, pad 1, groups=32 (4 ch/group) + ReLU.
// ---------------------------------------------------------------------------
__global__ __launch_bounds__(256) void k_gconv3(const float* __restrict__ t1,
                                                const float* __restrict__ Wt2,
                                                const float* __restrict__ bt2,
                                                float* __restrict__ t2) {
    int i = blockIdx.x * 256 + threadIdx.x;      // over B*128*NN
    int n = i & (NN - 1);
    int rest = i >> 11;
    int w = rest & (WD - 1);
    int b = rest >> 7;
    const float* base = t1 + ((size_t)b * WD + ((w >> 2) << 2)) * NN;
    float s = bt2[w];
#pragma unroll
    for (int j = 0; j < 4; ++j) {
        const float* row = base + (size_t)j * NN;
        const float* wp = Wt2 + (w * 4 + j) * 3;
        if (n > 0) s = fmaf(wp[0], row[n - 1], s);
        s = fmaf(wp[1], row[n], s);
        if (n < NN - 1) s = fmaf(wp[2], row[n + 1], s);
    }
    t2[((size_t)b * WD + w) * NN + n] = fmaxf(s, 0.f);
}

// ---------------------------------------------------------------------------
// KNN: block = 4 waves = 64 m-rows. A fragments held entirely in registers
// (128 VGPRs, loaded once). n-tiles staged transposed into double-buffered
// LDS by async global->LDS loads, pipelined one tile ahead
// (s_wait_asynccnt 32 = previous tile landed, in-order completion).
// Inner loop: 1 ds_load_b64 + 1 WMMA. Top-3 per lane-subset, LDS merge with
// (value desc, index asc) tie-break to match lax.top_k.
// ---------------------------------------------------------------------------
#define MR 64
#define LST 260

__global__ __launch_bounds__(128) void k_knn(const float* __restrict__ x,
                                             const float* __restrict__ sq,
                                             int* __restrict__ knn) {
    __shared__ float bnT[2][16][LST];            // 33,280 B, double-buffered

    int tid = threadIdx.x;
    int wave = tid >> 5, lane = tid & 31;
    int hi = lane >> 4, ln = lane & 15;
    int b = blockIdx.x >> 5, mt = blockIdx.x & 31;
    int m0 = mt * MR;
    const float* xb = x + (size_t)b * CCH * NN;
    const float* sqb = sq + (size_t)b * NN;

    // ---- A fragments in registers: af[kk] = x[4kk+2hi .. +1][m0+16w+ln] ----
    v2f af[64];
    {
        const float* amp = xb + m0 + 16 * wave + ln;
#pragma unroll
        for (int kk = 0; kk < 64; ++kk) {
            size_t r0 = (size_t)(4 * kk + 2 * hi) * NN;
            af[kk].x = amp[r0];
            af[kk].y = amp[r0 + NN];
        }
    }

    float v0[8], v1[8], v2[8], smv[8];
    int i0[8], i1[8], i2[8];
#pragma unroll
    for (int v = 0; v < 8; ++v) {
        v0[v] = v1[v] = v2[v] = -3.4e38f;
        i0[v] = i1[v] = i2[v] = 0;
        smv[v] = sqb[m0 + 16 * wave + v + 8 * hi];
    }

    // stage helper mapping: c = tid&15, k rows tid>>4, +8 => 32 loads/thread
    int sc = tid & 15, sk = tid >> 4;
    // ---- prologue: stage tile 0 into buf 0 ----
    for (int k = sk; k < CCH; k += 8) {
#if USE_ASYNC
        unsigned lds = (unsigned)(unsigned long long)&bnT[0][sc][k];
        unsigned go = (unsigned)((k * NN + 0 + sc) * 4);
        ASYNC_STAGE(lds, go, xb);
#else
        bnT[0][sc][k] = xb[(size_t)k * NN + sc];
#endif
    }

    for (int t = 0; t < NN / 16; ++t) {
        int n0 = t * 16;
        if (t + 1 < NN / 16) {                   // stage next tile (uniform)
            int nb = (t + 1) & 1;
            int nn0 = n0 + 16;
            for (int k = sk; k < CCH; k += 8) {
#if USE_ASYNC
                unsigned lds = (unsigned)(unsigned long long)&bnT[nb][sc][k];
                unsigned go = (unsigned)((k * NN + nn0 + sc) * 4);
                ASYNC_STAGE(lds, go, xb);
#else
                bnT[nb][sc][k] = xb[(size_t)k * NN + nn0 + sc];
#endif
            }
#if USE_ASYNC
            ASYNC_WAIT_PREV();                   // tile t landed (in-order)
#endif
        } else {
#if USE_ASYNC
            ASYNC_WAIT_ALL();
#endif
        }
        __syncthreads();                         // all waves' tile-t data ready

        const float* brow = bnT[t & 1][ln];
        v8f acc = {};
#pragma unroll
        for (int kk = 0; kk < 64; ++kk) {
            v2f bf = *(const v2f*)(brow + 4 * kk + 2 * hi);
            acc = wmma_f32_k4(af[kk], bf, acc);
        }
        float sn = sqb[n0 + ln];
        int nc = n0 + ln;
#pragma unroll
        for (int v = 0; v < 8; ++v) {
            float pd = 2.f * acc[v] - smv[v] - sn;
            if (pd > v0[v]) {
                v2[v] = v1[v]; i2[v] = i1[v];
                v1[v] = v0[v]; i1[v] = i0[v];
                v0[v] = pd;    i0[v] = nc;
            } else if (pd > v1[v]) {
                v2[v] = v1[v]; i2[v] = i1[v];
                v1[v] = pd;    i1[v] = nc;
            } else if (pd > v2[v]) {
                v2[v] = pd;    i2[v] = nc;
            }
        }
        __syncthreads();     // compute done before buf (t&1) restaged at t+2
    }

    // ---- merge 16 lane-subsets per row (overlay staging buffers) ----
    float* lvb = (float*)bnT;                    // [64][16][3] floats
    int* lib = (int*)bnT + MR * 48;              // [64][16][3] ints
#pragma unroll
    for (int v = 0; v < 8; ++v) {
        int rb = 16 * wave + v + 8 * hi;
        int base = (rb * 16 + ln) * 3;
        lvb[base + 0] = v0[v]; lib[base + 0] = i0[v];
        lvb[base + 1] = v1[v]; lib[base + 1] = i1[v];
        lvb[base + 2] = v2[v]; lib[base + 2] = i2[v];
    }
    __syncthreads();
    if (tid < MR) {
        float t0 = -3.4e38f, t1 = -3.4e38f, t2 = -3.4e38f;
        int j0 = 0x7fffffff, j1 = 0x7fffffff, j2 = 0x7fffffff;
        for (int l = 0; l < 16; ++l) {
#pragma unroll
            for (int s = 0; s < 3; ++s) {
                float vv = lvb[(tid * 16 + l) * 3 + s];
                int ii = lib[(tid * 16 + l) * 3 + s];
                bool g0 = (vv > t0) || (vv == t0 && ii < j0);
                bool g1 = (vv > t1) || (vv == t1 && ii < j1);
                bool g2 = (vv > t2) || (vv == t2 && ii < j2);
                if (g0) { t2 = t1; j2 = j1; t1 = t0; j1 = j0; t0 = vv; j0 = ii; }
                else if (g1) { t2 = t1; j2 = j1; t1 = vv; j1 = ii; }
                else if (g2) { t2 = vv; j2 = ii; }
            }
        }
        int* op = knn + ((size_t)b * NN + m0 + tid) * 3;
        op[0] = j0; op[1] = j1; op[2] = j2;
    }
}

// ---------------------------------------------------------------------------
// Fused semantic branch. Block = 8 waves handles one (b, 16-n) tile.
// GEMM1: per wave one m-tile, 3 col-tiles share one A fragment per k-step.
// GEMM3: 3 k-slices share one A fragment; B fragments are ds_load_b64.
// ---------------------------------------------------------------------------
__global__ __launch_bounds__(256) void k_semantic(
    const float* __restrict__ x, const float* __restrict__ Ws1,
    const float* __restrict__ bs1, const float* __restrict__ Ws2,
    const float* __restrict__ bs2, const float* __restrict__ Ws3,
    const float* __restrict__ bs3, const float* __restrict__ tout,
    const int* __restrict__ knn, float* __restrict__ out) {
    __shared__ int   sidx[48];
    __shared__ float s1[48][132];
    __shared__ float s2[48][132];

    int tid = threadIdx.x;
    int wave = tid >> 5, lane = tid & 31;
    int hi = lane >> 4, ln = lane & 15;
    int b = blockIdx.x >> 7, nt = blockIdx.x & 127;
    int n0 = nt * 16;
    const float* xb = x + (size_t)b * CCH * NN;

    if (tid < 48)
        sidx[tid] = knn[((size_t)b * NN + n0 + tid / 3) * 3 + (tid % 3)];
    __syncthreads();

    // --- GEMM1: s1 = relu(Ws1(128x512) * feat(512x48) + bs1) ---
    {
        int m0 = wave * 16;                       // 8 waves = 8 m-tiles
        int js0 = sidx[ln], js1 = sidx[16 + ln], js2 = sidx[32 + ln];
        int nc0 = n0 + ln / 3, nc1 = n0 + (16 + ln) / 3, nc2 = n0 + (32 + ln) / 3;
        const float* ap = Ws1 + (size_t)(m0 + ln) * 512 + 2 * hi;
        v8f a0 = {}, a1 = {}, a2 = {};
        for (int k = 0; k < 256; k += 4) {        // neighbor rows
            v2f a = *(const v2f*)(ap + k);
            size_t r0 = (size_t)(k + 2 * hi) * NN, r1 = r0 + NN;
            v2f f0, f1, f2;
            f0.x = xb[r0 + js0]; f0.y = xb[r1 + js0];
            f1.x = xb[r0 + js1]; f1.y = xb[r1 + js1];
            f2.x = xb[r0 + js2]; f2.y = xb[r1 + js2];
            a0 = wmma_f32_k4(a, f0, a0);
            a1 = wmma_f32_k4(a, f1, a1);
            a2 = wmma_f32_k4(a, f2, a2);
        }
        for (int k = 256; k < 512; k += 4) {      // center rows
            v2f a = *(const v2f*)(ap + k);
            size_t r0 = (size_t)(k - 256 + 2 * hi) * NN, r1 = r0 + NN;
            v2f f0, f1, f2;
            f0.x = xb[r0 + nc0]; f0.y = xb[r1 + nc0];
            f1.x = xb[r0 + nc1]; f1.y = xb[r1 + nc1];
            f2.x = xb[r0 + nc2]; f2.y = xb[r1 + nc2];
            a0 = wmma_f32_k4(a, f0, a0);
            a1 = wmma_f32_k4(a, f1, a1);
            a2 = wmma_f32_k4(a, f2, a2);
        }
#pragma unroll
        for (int v = 0; v < 8; ++v) {
            int m = m0 + v + 8 * hi;
            float bm = bs1[m];
            s1[ln][m]      = fmaxf(a0[v] + bm, 0.f);
            s1[16 + ln][m] = fmaxf(a1[v] + bm, 0.f);
            s1[32 + ln][m] = fmaxf(a2[v] + bm, 0.f);
        }
    }
    __syncthreads();

    // --- grouped 1x1 (4 ch/group) + ReLU -> s2 ---
    for (int i = tid; i < 48 * WD; i += 256) {
        int c = i >> 7, w = i & (WD - 1);
        int g4 = (w >> 2) << 2;
        float s = bs2[w];
#pragma unroll
        for (int j = 0; j < 4; ++j) s = fmaf(Ws2[w * 4 + j], s1[c][g4 + j], s);
        s2[c][w] = fmaxf(s, 0.f);
    }
    __syncthreads();

    // --- GEMM3: 3 k-slices share A; max over k; combine + residual ---
    const float* toutb = tout + (size_t)b * CCH * NN;
    float* outb = out + (size_t)b * CCH * NN;
    const float* sr0 = s2[ln * 3 + 0];
    const float* sr1 = s2[ln * 3 + 1];
    const float* sr2 = s2[ln * 3 + 2];
    for (int mt = wave; mt < 16; mt += 8) {
        int m0 = mt * 16;
        const float* ap = Ws3 + (size_t)(m0 + ln) * WD + 2 * hi;
        v8f c0 = {}, c1 = {}, c2 = {};
        for (int k = 0; k < WD; k += 4) {
            v2f a = *(const v2f*)(ap + k);
            v2f s0 = *(const v2f*)(sr0 + k + 2 * hi);
            v2f s1f = *(const v2f*)(sr1 + k + 2 * hi);
            v2f s2f = *(const v2f*)(sr2 + k + 2 * hi);
            c0 = wmma_f32_k4(a, s0, c0);
            c1 = wmma_f32_k4(a, s1f, c1);
            c2 = wmma_f32_k4(a, s2f, c2);
        }
        int n = n0 + ln;
#pragma unroll
        for (int v = 0; v < 8; ++v) {
            int m = m0 + v + 8 * hi;
            size_t o = (size_t)m * NN + n;
            float best = fmaxf(fmaxf(c0[v], c1[v]), c2[v]);
            outb[o] = fmaxf(best + bs3[m] + xb[o] + toutb[o], 0.f);
        }
    }
}

// ---------------------------------------------------------------------------
extern "C" void kernel_launch(void* const* d_in, const int* in_sizes, int n_in,
                              void* d_out, int out_size, void* d_ws,
                              size_t ws_size, hipStream_t stream) {
    const float* x   = (const float*)d_in[0];
    const float* Wt1 = (const float*)d_in[1];
    const float* bt1 = (const float*)d_in[2];
    const float* Wt2 = (const float*)d_in[3];
    const float* bt2 = (const float*)d_in[4];
    const float* Wt3 = (const float*)d_in[5];
    const float* bt3 = (const float*)d_in[6];
    const float* Ws1 = (const float*)d_in[7];
    const float* bs1 = (const float*)d_in[8];
    const float* Ws2 = (const float*)d_in[9];
    const float* bs2 = (const float*)d_in[10];
    const float* Ws3 = (const float*)d_in[11];
    const float* bs3 = (const float*)d_in[12];
    float* out = (float*)d_out;

    float* ws   = (float*)d_ws;
    float* sq   = ws;                                    // B*N
    float* t1   = sq + (size_t)BSZ * NN;                 // B*128*N
    float* t2   = t1 + (size_t)BSZ * WD * NN;            // B*128*N
    float* tout = t2 + (size_t)BSZ * WD * NN;            // B*256*N
    int*   idx  = (int*)(tout + (size_t)BSZ * CCH * NN); // B*N*3

    k_sqnorm<<<BSZ * NN / 256, 256, 0, stream>>>(x, sq);
    k_gemm<WD, CCH, 1><<<BSZ * (WD / 16) * (NN / 64) / 8, 256, 0, stream>>>(
        Wt1, bt1, x, t1, CCH * NN, WD * NN);
    k_gconv3<<<BSZ * WD * NN / 256, 256, 0, stream>>>(t1, Wt2, bt2, t2);
    k_gemm<CCH, WD, 0><<<BSZ * (CCH / 16) * (NN / 64) / 8, 256, 0, stream>>>(
        Wt3, bt3, t2, tout, WD * NN, CCH * NN);
    k_knn<<<BSZ * (NN / MR), 128, 0, stream>>>(x, sq, idx);
    k_semantic<<<BSZ * NN / 16, 256, 0, stream>>>(x, Ws1, bs1, Ws2, bs2, Ws3,
                                                  bs3, tout, idx, out);
}